// GIN_30227979829562
// MI455X (gfx1250) — compile-verified
//
#include <hip/hip_runtime.h>
#include <hip/hip_bf16.h>
#include <math.h>

// ---------------------------------------------------------------------------
// GIN forward for gfx1250 (MI455X). FP32 WMMA (V_WMMA_F32_16X16X4_F32) for all
// dense GEMMs; memory-bound scatter-add aggregation via global_atomic_add_f32
// (unsafeAtomicAdd => guaranteed HW FP atomic, no CAS loop).
// ---------------------------------------------------------------------------

typedef float v2f __attribute__((ext_vector_type(2)));
typedef float v8f __attribute__((ext_vector_type(8)));

// Device-pass-only sanity check (__HIP_DEVICE_COMPILE__ is defined by clang
// exclusively during device compilation; host pass must never see this).
#if defined(__HIP_DEVICE_COMPILE__)
#if !__has_builtin(__builtin_amdgcn_wmma_f32_16x16x4_f32)
#error "missing __builtin_amdgcn_wmma_f32_16x16x4_f32 on this toolchain (device pass)"
#endif
#endif

// Wrapper so the host compilation pass never parses the target builtin.
__device__ __forceinline__ v8f wmma_16x16x4_f32(v2f a, v2f b, v8f c) {
#if defined(__HIP_DEVICE_COMPILE__)
  return __builtin_amdgcn_wmma_f32_16x16x4_f32(
      /*neg_a=*/false, a, /*neg_b=*/false, b,
      /*c_mod=*/(short)0, c, /*reuse_a=*/false, /*reuse_b=*/false);
#else
  (void)a; (void)b;
  return c;  // host-pass placeholder; never executed
#endif
}

#define HID 128
#define NCLS 40
#define NPAD 48
#define STATS_BLOCKS 256

// ---------------- utility -------------------------------------------------
__global__ __launch_bounds__(256) void k_zero(float* __restrict__ p, size_t n) {
  size_t i = (size_t)blockIdx.x * 256 + threadIdx.x;
  if (i < n) p[i] = 0.0f;
}

// ---------------- scatter-add aggregation ---------------------------------
// 32 threads per edge; each thread moves a float4 (global_load_b128) and
// issues 4 hardware f32 atomics: agg[dst][c..c+3] += h[src][c..c+3].
__global__ __launch_bounds__(256) void k_scatter(
    const float* __restrict__ h, const int* __restrict__ src,
    const int* __restrict__ dst, float* __restrict__ agg, int nEdges) {
  size_t idx = (size_t)blockIdx.x * 256 + threadIdx.x;
  int e = (int)(idx >> 5);
  int c = (int)(idx & 31) * 4;
  if (e < nEdges) {
    int s = src[e];
    int d = dst[e];
    const float4 v = *(const float4*)(h + (size_t)s * HID + c);
    float* __restrict__ p = agg + (size_t)d * HID + c;
    unsafeAtomicAdd(p + 0, v.x);
    unsafeAtomicAdd(p + 1, v.y);
    unsafeAtomicAdd(p + 2, v.z);
    unsafeAtomicAdd(p + 3, v.w);
  }
}

// ---------------- fused-prologue WMMA GEMM (M x 128 x 128) ----------------
// MODE 0: in = A                      (head lin1)
// MODE 1: in = (1+eps[layer])*A + Agg (GIN combine, gemm1)
// MODE 2: in = A*scale + shift        (BN applied on the fly, gemm2)
// Y = relu(in @ W + bias).  Block = 256 threads = 8 waves; block owns 16 rows,
// wave w owns columns [16w, 16w+16).  K loop: 32 x v_wmma_f32_16x16x4_f32.
template <int MODE>
__global__ __launch_bounds__(256) void k_gemm(
    const float* __restrict__ A, const float* __restrict__ Agg,
    const float* __restrict__ scl, const float* __restrict__ shf,
    const float* __restrict__ epsArr, int layer,
    const float* __restrict__ W, const float* __restrict__ bias,
    float* __restrict__ Y) {
  const int lane = threadIdx.x & 31;
  const int wave = threadIdx.x >> 5;        // 0..7 -> N tile
  const int half = lane >> 4;               // 0/1: K-pair select (ISA A layout)
  const int l16  = lane & 15;
  const int m_base = blockIdx.x * 16;
  const int n_base = wave * 16;
  const int m = m_base + l16;               // A row for this lane
  const int n = n_base + l16;               // B/C/D column for this lane

  const float* __restrict__ arow = A + (size_t)m * HID;
  const float* __restrict__ grow = (MODE == 1) ? (Agg + (size_t)m * HID) : nullptr;
  float epc = 0.0f;
  if (MODE == 1) epc = 1.0f + epsArr[layer];

  v8f acc;
  {
    float bv = bias[n];
    #pragma unroll
    for (int r = 0; r < 8; ++r) acc[r] = bv;
  }

  const int koff = 2 * half;
  #pragma unroll 4
  for (int k = 0; k < HID; k += 4) {
    const int ka = k + koff;
    v2f a, b;
    if (MODE == 0) {
      a.x = arow[ka];
      a.y = arow[ka + 1];
    } else if (MODE == 1) {
      a.x = fmaf(epc, arow[ka],     grow[ka]);
      a.y = fmaf(epc, arow[ka + 1], grow[ka + 1]);
    } else {
      a.x = fmaf(arow[ka],     scl[ka],     shf[ka]);
      a.y = fmaf(arow[ka + 1], scl[ka + 1], shf[ka + 1]);
    }
    const float* __restrict__ bp = W + (size_t)ka * HID + n;
    b.x = bp[0];
    b.y = bp[HID];
    acc = wmma_16x16x4_f32(a, b, acc);
  }

  // C/D layout: lane<16 -> rows m_base+r, lane>=16 -> rows m_base+r+8
  float* __restrict__ op = Y + (size_t)(m_base + 8 * half) * HID + n;
  #pragma unroll
  for (int r = 0; r < 8; ++r) {
    float v = acc[r];
    op[(size_t)r * HID] = v > 0.0f ? v : 0.0f;   // ReLU
  }
}

// ---------------- BN column statistics (deterministic two-pass) -----------
__global__ __launch_bounds__(256) void k_colstats(
    const float* __restrict__ y, float* __restrict__ psum,
    float* __restrict__ psq, int M) {
  const int col = threadIdx.x & 127;
  const int rg  = threadIdx.x >> 7;   // 0/1
  float s = 0.0f, q = 0.0f;
  for (int row = blockIdx.x * 2 + rg; row < M; row += gridDim.x * 2) {
    float v = y[(size_t)row * HID + col];
    s += v;
    q += v * v;
  }
  __shared__ float sh_s[HID], sh_q[HID];
  if (rg == 0) { sh_s[col] = s; sh_q[col] = q; }
  __syncthreads();
  if (rg == 1) { sh_s[col] += s; sh_q[col] += q; }
  __syncthreads();
  if (rg == 1) {
    psum[(size_t)blockIdx.x * HID + col] = sh_s[col];
    psq [(size_t)blockIdx.x * HID + col] = sh_q[col];
  }
}

__global__ __launch_bounds__(128) void k_bnfinal(
    const float* __restrict__ psum, const float* __restrict__ psq,
    const float* __restrict__ gamma, const float* __restrict__ beta,
    float* __restrict__ scale, float* __restrict__ shift, float invM) {
  const int c = threadIdx.x;
  float s = 0.0f, q = 0.0f;
  for (int p = 0; p < STATS_BLOCKS; ++p) {
    s += psum[(size_t)p * HID + c];
    q += psq [(size_t)p * HID + c];
  }
  float mean = s * invM;
  float var  = q * invM - mean * mean;       // biased variance (matches ref)
  float sc   = gamma[c] * rsqrtf(var + 1e-5f);
  scale[c] = sc;
  shift[c] = beta[c] - mean * sc;
}

__global__ __launch_bounds__(256) void k_bnapply(
    const float* __restrict__ y, const float* __restrict__ scale,
    const float* __restrict__ shift, float* __restrict__ h, size_t total) {
  size_t i = (size_t)blockIdx.x * 256 + threadIdx.x;
  if (i < total) {
    int c = (int)(i & 127);
    h[i] = fmaf(y[i], scale[c], shift[c]);
  }
}

// ---------------- classifier head (N=40 padded to 48) ---------------------
__global__ __launch_bounds__(256) void k_padhead(
    const float* __restrict__ W, const float* __restrict__ b,
    float* __restrict__ Wp, float* __restrict__ bp) {
  int i = blockIdx.x * 256 + threadIdx.x;
  if (i < HID * NPAD) {
    int r = i / NPAD, c = i % NPAD;
    Wp[i] = (c < NCLS) ? W[r * NCLS + c] : 0.0f;
  }
  if (i < NPAD) bp[i] = (i < NCLS) ? b[i] : 0.0f;
}

__global__ __launch_bounds__(96) void k_gemmhead(
    const float* __restrict__ A, const float* __restrict__ Wp,
    const float* __restrict__ bp, float* __restrict__ logits) {
  const int lane = threadIdx.x & 31;
  const int wave = threadIdx.x >> 5;        // 0..2 -> N tile (48 cols)
  const int half = lane >> 4;
  const int l16  = lane & 15;
  const int m_base = blockIdx.x * 16;
  const int n_base = wave * 16;
  const int m = m_base + l16;
  const int n = n_base + l16;

  const float* __restrict__ arow = A + (size_t)m * HID;
  v8f acc;
  {
    float bv = bp[n];
    #pragma unroll
    for (int r = 0; r < 8; ++r) acc[r] = bv;
  }
  const int koff = 2 * half;
  #pragma unroll 4
  for (int k = 0; k < HID; k += 4) {
    const int ka = k + koff;
    v2f a, b;
    a.x = arow[ka];
    a.y = arow[ka + 1];
    const float* __restrict__ wp = Wp + (size_t)ka * NPAD + n;
    b.x = wp[0];
    b.y = wp[NPAD];
    acc = wmma_16x16x4_f32(a, b, acc);
  }
  if (n < NCLS) {
    float* __restrict__ op = logits + (size_t)(m_base + 8 * half) * NCLS + n;
    #pragma unroll
    for (int r = 0; r < 8; ++r) op[(size_t)r * NCLS] = acc[r];
  }
}

__global__ __launch_bounds__(256) void k_logsoftmax(
    const float* __restrict__ logits, float* __restrict__ out, int M) {
  int row = blockIdx.x * 256 + threadIdx.x;
  if (row >= M) return;
  const float* __restrict__ p = logits + (size_t)row * NCLS;
  float mx = p[0];
  for (int j = 1; j < NCLS; ++j) mx = fmaxf(mx, p[j]);
  float s = 0.0f;
  for (int j = 0; j < NCLS; ++j) s += expf(p[j] - mx);
  float lse = mx + logf(s);
  float* __restrict__ o = out + (size_t)row * NCLS;
  for (int j = 0; j < NCLS; ++j) o[j] = p[j] - lse;
}

// ---------------------------------------------------------------------------
extern "C" void kernel_launch(void* const* d_in, const int* in_sizes, int n_in,
                              void* d_out, int out_size, void* d_ws,
                              size_t ws_size, hipStream_t stream) {
  const float* x     = (const float*)d_in[0];
  const int*   ei    = (const int*)d_in[1];
  const float* eps   = (const float*)d_in[2];
  const float* W1s   = (const float*)d_in[3];
  const float* b1s   = (const float*)d_in[4];
  const float* g1s   = (const float*)d_in[5];
  const float* bt1s  = (const float*)d_in[6];
  const float* W2s   = (const float*)d_in[7];
  const float* b2s   = (const float*)d_in[8];
  const float* g2s   = (const float*)d_in[9];
  const float* bt2s  = (const float*)d_in[10];
  const float* lin1W = (const float*)d_in[11];
  const float* lin1b = (const float*)d_in[12];
  const float* lin2W = (const float*)d_in[13];
  const float* lin2b = (const float*)d_in[14];

  const int M = in_sizes[0] / HID;   // 50000 (multiple of 16)
  const int E = in_sizes[1] / 2;     // 800000
  const int* src = ei;
  const int* dst = ei + E;

  const size_t feat = (size_t)M * HID;
  float* ws = (float*)d_ws;
  float* h      = ws;                      ws += feat;            // BN'd activations
  float* agg    = ws;                      ws += feat;            // agg / y2
  float* y1     = ws;                      ws += feat;            // y1 / head hidden
  float* logits = ws;                      ws += (size_t)M * NCLS;
  float* psum   = ws;                      ws += (size_t)STATS_BLOCKS * HID;
  float* psq    = ws;                      ws += (size_t)STATS_BLOCKS * HID;
  float* scale  = ws;                      ws += HID;
  float* shift  = ws;                      ws += HID;
  float* Wpad   = ws;                      ws += HID * NPAD;
  float* bpad   = ws;                      ws += NPAD;

  const int gemmBlocks = M / 16;
  const int scatBlocks = (int)(((size_t)E * 32 + 255) / 256);
  const int zeroBlocks = (int)((feat + 255) / 256);
  const int bnapBlocks = zeroBlocks;
  const float invM = 1.0f / (float)M;

  for (int l = 0; l < 3; ++l) {
    const float* hin = (l == 0) ? x : h;
    const float* W1 = W1s + (size_t)l * HID * HID;
    const float* W2 = W2s + (size_t)l * HID * HID;

    // aggregation: agg[dst] += hin[src]
    k_zero<<<zeroBlocks, 256, 0, stream>>>(agg, feat);
    k_scatter<<<scatBlocks, 256, 0, stream>>>(hin, src, dst, agg, E);

    // y1 = relu(((1+eps)*hin + agg) @ W1 + b1)
    k_gemm<1><<<gemmBlocks, 256, 0, stream>>>(hin, agg, nullptr, nullptr, eps,
                                              l, W1, b1s + l * HID, y1);
    k_colstats<<<STATS_BLOCKS, 256, 0, stream>>>(y1, psum, psq, M);
    k_bnfinal<<<1, 128, 0, stream>>>(psum, psq, g1s + l * HID, bt1s + l * HID,
                                     scale, shift, invM);

    // y2 = relu(BN(y1) @ W2 + b2)  (BN fused into A loads; y2 reuses agg)
    k_gemm<2><<<gemmBlocks, 256, 0, stream>>>(y1, nullptr, scale, shift,
                                              nullptr, 0, W2, b2s + l * HID,
                                              agg);
    k_colstats<<<STATS_BLOCKS, 256, 0, stream>>>(agg, psum, psq, M);
    k_bnfinal<<<1, 128, 0, stream>>>(psum, psq, g2s + l * HID, bt2s + l * HID,
                                     scale, shift, invM);
    k_bnapply<<<bnapBlocks, 256, 0, stream>>>(agg, scale, shift, h, feat);
  }

  // head: relu(h @ lin1W + lin1b) -> lin2 (padded to 48 cols) -> log_softmax
  k_gemm<0><<<gemmBlocks, 256, 0, stream>>>(h, nullptr, nullptr, nullptr,
                                            nullptr, 0, lin1W, lin1b, y1);
  k_padhead<<<(HID * NPAD + 255) / 256, 256, 0, stream>>>(lin2W, lin2b, Wpad,
                                                          bpad);
  k_gemmhead<<<gemmBlocks, 96, 0, stream>>>(y1, Wpad, bpad, logits);
  k_logsoftmax<<<(M + 255) / 256, 256, 0, stream>>>(logits, (float*)d_out, M);
}